// GAT_Layer_39453569581649
// MI455X (gfx1250) — compile-verified
//
#include <hip/hip_runtime.h>
#include <cstdint>
#include <cstddef>

// ---------------------------------------------------------------------------
// GAT layer for MI455X (gfx1250, wave32).
// feat = x@W via V_WMMA_F32_16X16X4_F32 (fp32-faithful matrix path),
// edge softmax via encoded-uint atomic max + f32 atomic adds,
// BN stats via LDS partial reduction.
// ---------------------------------------------------------------------------

constexpr int   Bc  = 4;
constexpr int   Nn  = 50000;
constexpr int   Ec  = 800000;
constexpr int   INc = 128;
constexpr int   Hc  = 4;
constexpr int   Dc  = 32;
constexpr int   HDc = 128;           // H*D
constexpr float GAT_SLOPE = 0.2f;
constexpr float ACT_SLOPE = 0.01f;
constexpr float BN_EPS    = 1e-5f;

typedef __attribute__((ext_vector_type(2))) float v2f;
typedef __attribute__((ext_vector_type(8))) float v8f;

__device__ __forceinline__ float leakyf(float x, float s) {
  return x >= 0.0f ? x : s * x;
}

// Order-preserving float -> uint encoding for atomicMax-based segment max.
__device__ __forceinline__ unsigned enc_ord(float f) {
  unsigned u = __float_as_uint(f);
  return (u & 0x80000000u) ? ~u : (u | 0x80000000u);
}
// Decode; anything non-finite (including the 0-init sentinel) -> 0.0f
// (this is exactly the reference's zero-in-degree guard).
__device__ __forceinline__ float dec_ord_guard(unsigned u) {
  unsigned v = (u & 0x80000000u) ? (u ^ 0x80000000u) : ~u;
  if ((v & 0x7F800000u) == 0x7F800000u) return 0.0f; // inf / nan -> 0
  return __uint_as_float(v);
}

// ---------------------------------------------------------------------------
// Kernel 1: feat[b,n,:] = xx[b,n,:] @ W   via WMMA f32 16x16x4.
// Block = 256 threads = 8 waves. Each wave owns one 16-wide column tile of
// the 128 output channels; the block owns one 16-row node tile.
// grid.x = B * (N/16) = 4 * 3125 = 12500.
// ---------------------------------------------------------------------------
__global__ void gat_gemm_feat(const float* __restrict__ xx,
                              const float* __restrict__ W,
                              float* __restrict__ feat) {
  constexpr int tilesPerB = Nn / 16;                // 3125 (exact)
  const int t    = blockIdx.x;
  const int b    = t / tilesPerB;
  const int row0 = (t % tilesPerB) * 16;
  const int wave = threadIdx.x >> 5;                // 0..7 -> column tile
  const int lane = threadIdx.x & 31;
  const int col0 = wave * 16;
  const int mn   = lane & 15;        // A row / B,C column within tile
  const int kh   = (lane >> 4) * 2;  // K sub-offset: lanes 0-15 -> {k,k+1}, 16-31 -> {k+2,k+3}

  const float* __restrict__ arow = xx + ((size_t)b * Nn + row0 + mn) * INc;
  const float* __restrict__ bcol = W + col0 + mn;

  v8f acc = {};
#pragma unroll
  for (int k = 0; k < INc; k += 4) {
    v2f a, bb;
    a.x  = arow[k + kh];
    a.y  = arow[k + kh + 1];
    bb.x = bcol[(size_t)(k + kh) * HDc];
    bb.y = bcol[(size_t)(k + kh + 1) * HDc];
    // (neg_a, A, neg_b, B, c_mod, C, reuse_a, reuse_b)
    acc = __builtin_amdgcn_wmma_f32_16x16x4_f32(false, a, false, bb,
                                                (short)0, acc, false, false);
  }

  // C/D layout: VGPR r -> M = r + 8*(lane>=16), N = lane&15
  float* __restrict__ outp = feat + ((size_t)b * Nn + row0) * HDc + col0 + mn;
  const int rbase = (lane >> 4) * 8;
#pragma unroll
  for (int r = 0; r < 8; ++r) {
    outp[(size_t)(rbase + r) * HDc] = acc[r];
  }
}

// ---------------------------------------------------------------------------
// Kernel 2: el/er[b,n,h] = <feat[b,n,h,:], attn_l/r[h,:]>
// one thread per (b,n,h); 800k threads.
// ---------------------------------------------------------------------------
__global__ void gat_attn_dots(const float* __restrict__ feat,
                              const float* __restrict__ attn_l,
                              const float* __restrict__ attn_r,
                              float* __restrict__ el,
                              float* __restrict__ er) {
  size_t i = (size_t)blockIdx.x * blockDim.x + threadIdx.x;
  if (i >= (size_t)Bc * Nn * Hc) return;
  const int    h  = (int)(i % Hc);
  const size_t bn = i / Hc;
  const float* __restrict__ f  = feat + bn * HDc + h * Dc;
  const float* __restrict__ al = attn_l + h * Dc;
  const float* __restrict__ ar = attn_r + h * Dc;
  float sl = 0.0f, sr = 0.0f;
#pragma unroll
  for (int d = 0; d < Dc; ++d) {
    const float v = f[d];
    sl = fmaf(v, al[d], sl);
    sr = fmaf(v, ar[d], sr);
  }
  el[i] = sl;
  er[i] = sr;
}

// ---------------------------------------------------------------------------
// Kernel 3: segment max of e = leaky(el[src]+er[dst]) over dst, per (b,h).
// one thread per (b,e); atomicMax on order-preserving encoding.
// ---------------------------------------------------------------------------
__global__ void gat_edge_max(const float* __restrict__ el,
                             const float* __restrict__ er,
                             const int* __restrict__ src,
                             const int* __restrict__ dst,
                             unsigned* __restrict__ mEnc) {
  size_t i = (size_t)blockIdx.x * blockDim.x + threadIdx.x;
  if (i >= (size_t)Bc * Ec) return;
  const int b = (int)(i / Ec);
  const int e = (int)(i % Ec);
  const int s = src[e];
  const int d = dst[e];
  const size_t sb = ((size_t)b * Nn + s) * Hc;
  const size_t db = ((size_t)b * Nn + d) * Hc;
#pragma unroll
  for (int h = 0; h < Hc; ++h) {
    const float ev = leakyf(el[sb + h] + er[db + h], GAT_SLOPE);
    atomicMax(&mEnc[db + h], enc_ord(ev));
  }
}

// ---------------------------------------------------------------------------
// Kernel 4: w = exp(e - m[dst]); denom[dst] += w.
// ---------------------------------------------------------------------------
__global__ void gat_edge_expsum(const float* __restrict__ el,
                                const float* __restrict__ er,
                                const int* __restrict__ src,
                                const int* __restrict__ dst,
                                const unsigned* __restrict__ mEnc,
                                float* __restrict__ denom) {
  size_t i = (size_t)blockIdx.x * blockDim.x + threadIdx.x;
  if (i >= (size_t)Bc * Ec) return;
  const int b = (int)(i / Ec);
  const int e = (int)(i % Ec);
  const int s = src[e];
  const int d = dst[e];
  const size_t sb = ((size_t)b * Nn + s) * Hc;
  const size_t db = ((size_t)b * Nn + d) * Hc;
#pragma unroll
  for (int h = 0; h < Hc; ++h) {
    const float ev = leakyf(el[sb + h] + er[db + h], GAT_SLOPE);
    const float md = dec_ord_guard(mEnc[db + h]);
    atomicAdd(&denom[db + h], __expf(ev - md));
  }
}

// ---------------------------------------------------------------------------
// Kernel 5: out[b,dst,h,:] += alpha * feat[b,src,h,:].
// One wave32 per (b,e); lane = d (perfectly coalesced 128B per head).
// ---------------------------------------------------------------------------
__global__ void gat_edge_aggregate(const float* __restrict__ feat,
                                   const float* __restrict__ el,
                                   const float* __restrict__ er,
                                   const int* __restrict__ src,
                                   const int* __restrict__ dst,
                                   const unsigned* __restrict__ mEnc,
                                   const float* __restrict__ denom,
                                   float* __restrict__ out) {
  const size_t wid = (size_t)blockIdx.x * (blockDim.x >> 5) + (threadIdx.x >> 5);
  if (wid >= (size_t)Bc * Ec) return;
  const int lane = threadIdx.x & 31;
  const int b = (int)(wid / Ec);
  const int e = (int)(wid % Ec);
  const int s = src[e];
  const int d = dst[e];
  const size_t sb = ((size_t)b * Nn + s) * Hc;
  const size_t db = ((size_t)b * Nn + d) * Hc;
  const float* __restrict__ fsrc = feat + ((size_t)b * Nn + s) * HDc;
  float* __restrict__ odst       = out + ((size_t)b * Nn + d) * HDc;
  __builtin_prefetch(fsrc, 0, 0);
#pragma unroll
  for (int h = 0; h < Hc; ++h) {
    const float ev    = leakyf(el[sb + h] + er[db + h], GAT_SLOPE);
    const float md    = dec_ord_guard(mEnc[db + h]);
    const float w     = __expf(ev - md);
    const float den   = fmaxf(denom[db + h], 1e-9f);
    const float alpha = w / den;
    atomicAdd(&odst[h * Dc + lane], alpha * fsrc[h * Dc + lane]);
  }
}

// ---------------------------------------------------------------------------
// Kernel 6: BN batch stats: per-channel sum / sumsq over B*N rows.
// Block = 256 = 2 row-lanes x 128 channels; 64 rows per block; LDS combine.
// grid.x = (B*N)/64 = 3125.
// ---------------------------------------------------------------------------
__global__ void gat_bn_stats(const float* __restrict__ h,
                             float* __restrict__ chanSum,
                             float* __restrict__ chanSumSq) {
  const int c  = threadIdx.x & 127;
  const int rl = threadIdx.x >> 7;               // 0 or 1
  const size_t row0 = (size_t)blockIdx.x * 64;
  const size_t totalRows = (size_t)Bc * Nn;
  float s = 0.0f, s2 = 0.0f;
  for (int i = rl; i < 64; i += 2) {
    const size_t r = row0 + i;
    if (r < totalRows) {
      const float v = h[r * HDc + c];
      s  += v;
      s2 = fmaf(v, v, s2);
    }
  }
  __shared__ float sh[128];
  __shared__ float sh2[128];
  if (rl == 0) { sh[c] = s; sh2[c] = s2; }
  __syncthreads();
  if (rl == 1) { sh[c] += s; sh2[c] += s2; }
  __syncthreads();
  if (rl == 0) {
    atomicAdd(&chanSum[c],   sh[c]);
    atomicAdd(&chanSumSq[c], sh2[c]);
  }
}

// ---------------------------------------------------------------------------
// Kernel 7: BN apply (training-mode batch stats) + LeakyReLU(0.01), in place.
// ---------------------------------------------------------------------------
__global__ void gat_bn_apply(float* __restrict__ out,
                             const float* __restrict__ chanSum,
                             const float* __restrict__ chanSumSq,
                             const float* __restrict__ gamma,
                             const float* __restrict__ beta) {
  const size_t i = (size_t)blockIdx.x * blockDim.x + threadIdx.x;
  if (i >= (size_t)Bc * Nn * HDc) return;
  const int c = (int)(i & (HDc - 1));
  const float invM = 1.0f / (float)((size_t)Bc * Nn);
  const float mean = chanSum[c] * invM;
  const float var  = chanSumSq[c] * invM - mean * mean;
  const float v    = out[i];
  const float hn   = (v - mean) * __frsqrt_rn(var + BN_EPS) * gamma[c] + beta[c];
  out[i] = leakyf(hn, ACT_SLOPE);
}

// ---------------------------------------------------------------------------
// Host-side launch.  Inputs (setup_inputs order):
//   0: xx [B,N,IN] f32     1: W [IN,H*D] f32    2: attn_l [H,D] f32
//   3: attn_r [H,D] f32    4: gamma [H*D] f32   5: beta [H*D] f32
//   6: src [E] i32         7: dst [E] i32
// Output: [B,N,H*D] f32.
// ---------------------------------------------------------------------------
extern "C" void kernel_launch(void* const* d_in, const int* in_sizes, int n_in,
                              void* d_out, int out_size, void* d_ws, size_t ws_size,
                              hipStream_t stream) {
  const float* xx     = (const float*)d_in[0];
  const float* W      = (const float*)d_in[1];
  const float* attn_l = (const float*)d_in[2];
  const float* attn_r = (const float*)d_in[3];
  const float* gamma  = (const float*)d_in[4];
  const float* beta   = (const float*)d_in[5];
  const int*   src    = (const int*)d_in[6];
  const int*   dst    = (const int*)d_in[7];
  float*       out    = (float*)d_out;

  // Workspace carve-up (bytes):
  const size_t featBytes = (size_t)Bc * Nn * HDc * sizeof(float);   // 102,400,000
  const size_t nhBytes   = (size_t)Bc * Nn * Hc * sizeof(float);    //   3,200,000
  char* ws = (char*)d_ws;
  float*    feat      = (float*)(ws);
  float*    el        = (float*)(ws + featBytes);
  float*    er        = (float*)(ws + featBytes + nhBytes);
  unsigned* mEnc      = (unsigned*)(ws + featBytes + 2 * nhBytes);
  float*    denom     = (float*)(ws + featBytes + 3 * nhBytes);
  float*    chanSum   = (float*)(ws + featBytes + 4 * nhBytes);
  float*    chanSumSq = (float*)(ws + featBytes + 4 * nhBytes + 512);
  // total ws use: featBytes + 4*nhBytes + 1024  (~115.2 MB)

  // Zero-init accumulators (mEnc==0 decodes to NaN -> zero-in-degree guard).
  hipMemsetAsync(out, 0, (size_t)out_size * sizeof(float), stream);
  hipMemsetAsync(mEnc, 0, nhBytes, stream);
  hipMemsetAsync(denom, 0, nhBytes, stream);
  hipMemsetAsync(chanSum, 0, 1024, stream);

  // 1) GEMM (WMMA fp32 path)
  gat_gemm_feat<<<Bc * (Nn / 16), 256, 0, stream>>>(xx, W, feat);

  // 2) attention dot products
  {
    const size_t n = (size_t)Bc * Nn * Hc;
    gat_attn_dots<<<(unsigned)((n + 255) / 256), 256, 0, stream>>>(feat, attn_l, attn_r, el, er);
  }

  // 3) segment max
  {
    const size_t n = (size_t)Bc * Ec;
    gat_edge_max<<<(unsigned)((n + 255) / 256), 256, 0, stream>>>(el, er, src, dst, mEnc);
  }

  // 4) exp + denom
  {
    const size_t n = (size_t)Bc * Ec;
    gat_edge_expsum<<<(unsigned)((n + 255) / 256), 256, 0, stream>>>(el, er, src, dst, mEnc, denom);
  }

  // 5) aggregation: one wave per (b, e)
  {
    const size_t waves = (size_t)Bc * Ec;           // 3.2M waves
    const unsigned blocks = (unsigned)((waves + 7) / 8);  // 8 waves / 256-thread block
    gat_edge_aggregate<<<blocks, 256, 0, stream>>>(feat, el, er, src, dst, mEnc, denom, out);
  }

  // 6) BN stats
  gat_bn_stats<<<(Bc * Nn) / 64, 256, 0, stream>>>(out, chanSum, chanSumSq);

  // 7) BN apply + LeakyReLU
  {
    const size_t n = (size_t)Bc * Nn * HDc;
    gat_bn_apply<<<(unsigned)((n + 255) / 256), 256, 0, stream>>>(out, chanSum, chanSumSq, gamma, beta);
  }
}